// CustomDecoupledS2GC_2877628089024
// MI455X (gfx1250) — compile-verified
//
#include <hip/hip_runtime.h>
#include <hip/hip_bf16.h>

#define GN 100000
#define GE 1250000
#define GD 64
#define GH 256
#define GC 40
#define GK 4
#define G_ALPHA 0.05f
#define G_EPS 1e-5f
#define G_C0 ((1.0f - G_ALPHA) / (float)GK)   // 0.2375

typedef float v2f __attribute__((ext_vector_type(2)));
typedef float v8f __attribute__((ext_vector_type(8)));

// ---------------- elementwise helpers ----------------

__global__ void k_fill(float* __restrict__ p, int n, float v) {
    int i = blockIdx.x * blockDim.x + threadIdx.x;
    if (i < n) p[i] = v;
}

__global__ void k_copy(float* __restrict__ d, const float* __restrict__ s, int n) {
    int i = blockIdx.x * blockDim.x + threadIdx.x;
    if (i < n) d[i] = s[i];
}

// ---------------- graph propagation ----------------

__global__ void k_deg(const int* __restrict__ dst, float* __restrict__ deg, int E) {
    int e = blockIdx.x * blockDim.x + threadIdx.x;
    if (e < E) unsafeAtomicAdd(&deg[dst[e]], 1.0f);
}

__global__ void k_norm(const float* __restrict__ deg, float* __restrict__ nrm, int n) {
    int i = blockIdx.x * blockDim.x + threadIdx.x;
    if (i < n) {
        float d = deg[i];
        d = d < 1.0f ? 1.0f : d;
        nrm[i] = rsqrtf(d);
    }
}

// h_next[dst, d] += h[src, d] * norm[src]   (pre-scale fused into gather)
__global__ void k_scatter(const float* __restrict__ h, const float* __restrict__ nrm,
                          const int* __restrict__ src, const int* __restrict__ dst,
                          float* __restrict__ hn, int E) {
    int gid = blockIdx.x * blockDim.x + threadIdx.x;
    if (gid >= E * GD) return;
    int e = gid >> 6;      // GD == 64
    int d = gid & 63;
    int s = src[e];
    int t = dst[e];
    float v = h[(size_t)s * GD + d] * nrm[s];
    unsafeAtomicAdd(&hn[(size_t)t * GD + d], v);
}

// v = h_next * norm ; h = v ; acc += v ; h_next = 0 (for next round)
__global__ void k_finalize(float* __restrict__ h, float* __restrict__ hn,
                           float* __restrict__ acc, const float* __restrict__ nrm, int n) {
    int gid = blockIdx.x * blockDim.x + threadIdx.x;
    if (gid >= n) return;
    int i = gid >> 6;
    float v = hn[gid] * nrm[i];
    hn[gid] = 0.0f;
    h[gid] = v;
    acc[gid] += v;
}

// X = (1-alpha)/K * acc + alpha * feat
__global__ void k_mix(float* __restrict__ X, const float* __restrict__ acc,
                      const float* __restrict__ feat, int n) {
    int i = blockIdx.x * blockDim.x + threadIdx.x;
    if (i < n) X[i] = G_C0 * acc[i] + G_ALPHA * feat[i];
}

// ---------------- WMMA GEMMs (V_WMMA_F32_16X16X4_F32, exact fp32) ----------------
// A: [N, K] row-major (lda = K), B: [K, 256] row-major, Y: [N, 256], bias added.
// block = 256 thr (8 waves): 2 row-tiles x 4 col-tiles of 16x16.
// grid = (N/32, 256/64)
__global__ void k_gemm_wmma(const float* __restrict__ A, const float* __restrict__ W,
                            const float* __restrict__ bias, float* __restrict__ Y,
                            int K, int lda) {
    int lane = threadIdx.x & 31;
    int wid  = threadIdx.x >> 5;
    int m0 = (blockIdx.x * 2 + (wid >> 2)) * 16;
    int n0 = (blockIdx.y * 4 + (wid & 3)) * 16;

    int rA   = m0 + (lane & 15);          // A row for this lane
    int colB = n0 + (lane & 15);          // B col for this lane
    int ksel = (lane >> 4) * 2;           // lanes 0-15 -> K+0,1 ; lanes 16-31 -> K+2,3

    v8f c = {};
    for (int k0 = 0; k0 < K; k0 += 4) {
        int kk = k0 + ksel;
        v2f a, b;
        a.x = A[(size_t)rA * lda + kk];
        a.y = A[(size_t)rA * lda + kk + 1];
        b.x = W[(size_t)kk * GH + colB];
        b.y = W[(size_t)(kk + 1) * GH + colB];
        c = __builtin_amdgcn_wmma_f32_16x16x4_f32(false, a, false, b, (short)0, c, false, false);
    }

    int rowBase = m0 + ((lane >> 4) * 8); // C layout: VGPR r -> M = r (lanes<16) / r+8
    int col = n0 + (lane & 15);
    float bv = bias[col];
#pragma unroll
    for (int r = 0; r < 8; ++r)
        Y[(size_t)(rowBase + r) * GH + col] = c[r] + bv;
}

// Final layer: A [N,256] @ W3 [256,40] + b3 -> out [N,40].
// block = 96 thr (3 waves = 3 col-tiles covering cols 0..47, cols>=40 masked).
// grid = N/16
__global__ void k_gemm3_wmma(const float* __restrict__ A, const float* __restrict__ W,
                             const float* __restrict__ bias, float* __restrict__ out) {
    int lane = threadIdx.x & 31;
    int wid  = threadIdx.x >> 5;          // 0..2
    int m0 = blockIdx.x * 16;
    int n0 = wid * 16;

    int rA  = m0 + (lane & 15);
    int col = n0 + (lane & 15);
    bool vcol = (col < GC);
    int ksel = (lane >> 4) * 2;

    v8f c = {};
    for (int k0 = 0; k0 < GH; k0 += 4) {
        int kk = k0 + ksel;
        v2f a, b;
        a.x = A[(size_t)rA * GH + kk];
        a.y = A[(size_t)rA * GH + kk + 1];
        b.x = vcol ? W[(size_t)kk * GC + col] : 0.0f;
        b.y = vcol ? W[(size_t)(kk + 1) * GC + col] : 0.0f;
        c = __builtin_amdgcn_wmma_f32_16x16x4_f32(false, a, false, b, (short)0, c, false, false);
    }

    if (vcol) {
        float bv = bias[col];
        int rowBase = m0 + ((lane >> 4) * 8);
#pragma unroll
        for (int r = 0; r < 8; ++r)
            out[(size_t)(rowBase + r) * GC + col] = c[r] + bv;
    }
}

// ---------------- batchnorm (column stats over N rows) ----------------

#define STAT_ROWS 256
__global__ void k_colstats(const float* __restrict__ Y, float* __restrict__ stats, int N) {
    int col = threadIdx.x;                // 256 threads = 256 columns
    int r0 = blockIdx.x * STAT_ROWS;
    int r1 = r0 + STAT_ROWS; if (r1 > N) r1 = N;
    float s = 0.0f, s2 = 0.0f;
    for (int r = r0; r < r1; ++r) {
        float v = Y[(size_t)r * GH + col];
        s += v; s2 += v * v;
    }
    unsafeAtomicAdd(&stats[col], s);
    unsafeAtomicAdd(&stats[GH + col], s2);
}

__global__ void k_bnparams(const float* __restrict__ stats, const float* __restrict__ g,
                           const float* __restrict__ be, float* __restrict__ params) {
    int c = threadIdx.x;
    const float invN = 1.0f / (float)GN;
    float mean = stats[c] * invN;
    float var  = stats[GH + c] * invN - mean * mean;
    float sc = g[c] * rsqrtf(var + G_EPS);
    params[c] = sc;
    params[GH + c] = be[c] - mean * sc;
}

__global__ void k_bnrelu(float* __restrict__ Y, const float* __restrict__ params, int n) {
    int i = blockIdx.x * blockDim.x + threadIdx.x;
    if (i >= n) return;
    int c = i & (GH - 1);
    float v = Y[i] * params[c] + params[GH + c];
    Y[i] = v > 0.0f ? v : 0.0f;
}

// ---------------- launch ----------------

extern "C" void kernel_launch(void* const* d_in, const int* in_sizes, int n_in,
                              void* d_out, int out_size, void* d_ws, size_t ws_size,
                              hipStream_t stream) {
    const float* feat = (const float*)d_in[0];
    const int*   src  = (const int*)d_in[1];
    const int*   dstE = (const int*)d_in[2];
    const float* W1   = (const float*)d_in[3];
    const float* b1   = (const float*)d_in[4];
    const float* g1   = (const float*)d_in[5];
    const float* be1  = (const float*)d_in[6];
    const float* W2   = (const float*)d_in[7];
    const float* b2   = (const float*)d_in[8];
    const float* g2   = (const float*)d_in[9];
    const float* be2  = (const float*)d_in[10];
    const float* W3   = (const float*)d_in[11];
    const float* b3   = (const float*)d_in[12];
    float* out = (float*)d_out;

    float* ws = (float*)d_ws;
    const size_t ND = (size_t)GN * GD;    // 6.4M
    const size_t NH = (size_t)GN * GH;    // 25.6M
    float* h     = ws;                    // [N,D]   h, later X
    float* acc   = h + ND;                // [N,D]
    float* hn    = acc + ND;              // [N,D] h_next; region later reused as Y
    float* Y     = hn;                    // [N,H]
    float* Z     = Y + NH;                // [N,H]
    float* deg   = Z + NH;                // [N]
    float* nrm   = deg + GN;              // [N]
    float* stats = nrm + GN;              // [2H]
    float* params= stats + 2 * GH;        // [2H]

    const int T = 256;
    auto blocks = [](long n, int t) { return (int)((n + t - 1) / t); };

    // init
    k_fill<<<blocks(GN, T), T, 0, stream>>>(deg, GN, 0.0f);
    k_fill<<<blocks(ND, T), T, 0, stream>>>(acc, (int)ND, 0.0f);
    k_fill<<<blocks(ND, T), T, 0, stream>>>(hn,  (int)ND, 0.0f);
    k_copy<<<blocks(ND, T), T, 0, stream>>>(h, feat, (int)ND);

    // degree + norm
    k_deg<<<blocks(GE, T), T, 0, stream>>>(dstE, deg, GE);
    k_norm<<<blocks(GN, T), T, 0, stream>>>(deg, nrm, GN);

    // K propagation rounds
    const long EW = (long)GE * GD;        // 80M work items
    for (int k = 0; k < GK; ++k) {
        k_scatter<<<blocks(EW, T), T, 0, stream>>>(h, nrm, src, dstE, hn, GE);
        k_finalize<<<blocks(ND, T), T, 0, stream>>>(h, hn, acc, nrm, (int)ND);
    }

    // X = c0*acc + alpha*feat  (into h)
    k_mix<<<blocks(ND, T), T, 0, stream>>>(h, acc, feat, (int)ND);

    // layer 1: [N,64] @ [64,256] + b1 -> Y, then BN+ReLU
    {
        dim3 grid(GN / 32, GH / 64);
        k_gemm_wmma<<<grid, 256, 0, stream>>>(h, W1, b1, Y, GD, GD);
        k_fill<<<1, 2 * GH, 0, stream>>>(stats, 2 * GH, 0.0f);
        k_colstats<<<blocks(GN, STAT_ROWS), GH, 0, stream>>>(Y, stats, GN);
        k_bnparams<<<1, GH, 0, stream>>>(stats, g1, be1, params);
        k_bnrelu<<<blocks(NH, T), T, 0, stream>>>(Y, params, (int)NH);
    }

    // layer 2: [N,256] @ [256,256] + b2 -> Z, then BN+ReLU
    {
        dim3 grid(GN / 32, GH / 64);
        k_gemm_wmma<<<grid, 256, 0, stream>>>(Y, W2, b2, Z, GH, GH);
        k_fill<<<1, 2 * GH, 0, stream>>>(stats, 2 * GH, 0.0f);
        k_colstats<<<blocks(GN, STAT_ROWS), GH, 0, stream>>>(Z, stats, GN);
        k_bnparams<<<1, GH, 0, stream>>>(stats, g2, be2, params);
        k_bnrelu<<<blocks(NH, T), T, 0, stream>>>(Z, params, (int)NH);
    }

    // layer 3: [N,256] @ [256,40] + b3 -> out
    k_gemm3_wmma<<<GN / 16, 96, 0, stream>>>(Z, W3, b3, out);
}